// LatentVQVAE_49598282334508
// MI455X (gfx1250) — compile-verified
//
#include <hip/hip_runtime.h>
#include <math.h>

#define B_N   512
#define H_N   14
#define W_N   14
#define DIN   512
#define EDIM  64
#define E1    128
#define K_CB  256
#define NTOK  2048
#define NPIX  49

#define OFF_PERP   51380224      // B*H*W*DIN
#define OFF_COMMIT (OFF_PERP + 1)
#define OFF_IDX    (OFF_PERP + 2)

typedef __attribute__((ext_vector_type(2))) float v2f;
typedef __attribute__((ext_vector_type(4))) float v4f;
typedef __attribute__((ext_vector_type(8))) float v8f;

static __device__ __forceinline__ v8f wmma_k4(v2f a, v2f b, v8f c) {
  // D = A(16x4,f32) * B(4x16,f32) + C(16x16,f32), exact fp32
  return __builtin_amdgcn_wmma_f32_16x16x4_f32(false, a, false, b, (short)0, c,
                                               false, false);
}

static __device__ __forceinline__ float gelu_exact(float x) {
  return 0.5f * x * (1.0f + erff(x * 0.70710678118654752440f));
}

#define SDELTA_STRIDE 513   // 513 % 64 == 1 -> conflict-free A-frag reads
#define SH_STRIDE     129
#define SQ_STRIDE     65

// ---------------------------------------------------------------------------
// Kernel 1: fused delta -> enc layer1(GELU) -> enc layer2 -> 7x7 avg pool.
// One workgroup per (b, gy, gx) token.  8 waves, each owns a 16-col tile of
// layer1 and accumulates 4 row tiles so enc_w1 is streamed exactly once/WG.
// ---------------------------------------------------------------------------
__global__ __launch_bounds__(256) void vqvae_encode_pool(
    const float* __restrict__ t0, const float* __restrict__ t1,
    const float* __restrict__ w1, const float* __restrict__ b1,
    const float* __restrict__ w2, const float* __restrict__ b2,
    float* __restrict__ enc_out)
{
  __shared__ float smem[64 * SDELTA_STRIDE];  // delta tile, later reused for h
  __shared__ float sPool[EDIM];

  const int tid  = threadIdx.x;
  const int wave = tid >> 5;
  const int lane = tid & 31;
  const int blk  = blockIdx.x & 3;
  const int b    = blockIdx.x >> 2;
  const int gy   = blk >> 1, gx = blk & 1;

  if (tid < EDIM) sPool[tid] = 0.0f;

  // Stage delta: 49 valid rows (7x7 pixels), rows 49..63 zero-padded.
  for (int i = 0; i < 32; ++i) {
    int e = tid + (i << 8);            // 0..8191 : 64 rows * 128 float4
    int r = e >> 7;
    int c = (e & 127) << 2;
    v4f d = (v4f){0.f, 0.f, 0.f, 0.f};
    if (r < NPIX) {
      int ph = gy * 7 + r / 7;
      int pw = gx * 7 + r % 7;
      size_t base = (((size_t)b * H_N + ph) * W_N + pw) * (size_t)DIN + c;
      v4f a1 = __builtin_nontemporal_load((const v4f*)(t1 + base));
      v4f a0 = __builtin_nontemporal_load((const v4f*)(t0 + base));
      d = a1 - a0;
    }
    float* dst = &smem[r * SDELTA_STRIDE + c];
    dst[0] = d.x; dst[1] = d.y; dst[2] = d.z; dst[3] = d.w;
  }
  __syncthreads();

  const int n    = lane & 15;
  const int khi  = (lane >> 4) << 1;   // 0 (lanes 0-15) or 2 (lanes 16-31)
  const int hi8  = (lane >> 4) << 3;   // C-tile row offset 0 / 8
  const int col1 = (wave << 4) + n;    // this wave's layer1 column

  v8f acc[4];
  #pragma unroll
  for (int t = 0; t < 4; ++t)
    acc[t] = (v8f){0.f,0.f,0.f,0.f,0.f,0.f,0.f,0.f};

  // Layer 1: delta[64x512] @ w1[512x128]
  for (int kk = 0; kk < DIN; kk += 4) {
    const int k0 = kk + khi;
    v2f bf;
    bf.x = w1[k0 * E1 + col1];
    bf.y = w1[(k0 + 1) * E1 + col1];
    #pragma unroll
    for (int t = 0; t < 4; ++t) {
      v2f af;
      af.x = smem[(t * 16 + n) * SDELTA_STRIDE + k0];
      af.y = smem[(t * 16 + n) * SDELTA_STRIDE + k0 + 1];
      acc[t] = wmma_k4(af, bf, acc[t]);
    }
  }

  const float bias1 = b1[col1];
  __syncthreads();                     // all delta reads done before overwrite
  #pragma unroll
  for (int t = 0; t < 4; ++t)
    #pragma unroll
    for (int i = 0; i < 8; ++i)
      smem[(t * 16 + hi8 + i) * SH_STRIDE + col1] =
          gelu_exact(acc[t][i] + bias1);
  __syncthreads();

  // Layer 2: h[64x128] @ w2[128x64] -> 16 tiles, 2 per wave; pool rows < 49.
  #pragma unroll
  for (int t = 0; t < 2; ++t) {
    const int id = wave * 2 + t;
    const int mt = id >> 2;
    const int ct = id & 3;
    const int col2 = (ct << 4) + n;
    v8f a2 = (v8f){0.f,0.f,0.f,0.f,0.f,0.f,0.f,0.f};
    for (int kk = 0; kk < E1; kk += 4) {
      const int k0 = kk + khi;
      v2f bf;
      bf.x = w2[k0 * EDIM + col2];
      bf.y = w2[(k0 + 1) * EDIM + col2];
      v2f af;
      af.x = smem[(mt * 16 + n) * SH_STRIDE + k0];
      af.y = smem[(mt * 16 + n) * SH_STRIDE + k0 + 1];
      a2 = wmma_k4(af, bf, a2);
    }
    const float bias2 = b2[col2];
    float part = 0.f;
    #pragma unroll
    for (int i = 0; i < 8; ++i) {
      int row = mt * 16 + hi8 + i;
      if (row < NPIX) part += a2[i] + bias2;
    }
    atomicAdd(&sPool[col2], part);
  }
  __syncthreads();
  if (tid < EDIM)
    enc_out[(size_t)blockIdx.x * EDIM + tid] = sPool[tid] * (1.0f / 49.0f);
}

// ---------------------------------------------------------------------------
// Kernel 2: nearest-code search per token + commitment partial + histogram.
// One block per token; thread tid scores code tid (K == 256 == blockDim).
// ---------------------------------------------------------------------------
__global__ __launch_bounds__(256) void vqvae_vq(
    const float* __restrict__ enc, const float* __restrict__ codebook,
    int* __restrict__ idx_out, float* __restrict__ commit_accum,
    int* __restrict__ hist, float* __restrict__ out_indices)
{
  __shared__ float sF[EDIM];
  __shared__ float sScore[256];
  __shared__ int   sIdx[256];
  const int t = blockIdx.x, tid = threadIdx.x;
  if (tid < EDIM) sF[tid] = enc[(size_t)t * EDIM + tid];
  __syncthreads();

  const float* cb = codebook + (size_t)tid * EDIM;
  float dot = 0.f, c2 = 0.f;
  #pragma unroll 8
  for (int d = 0; d < EDIM; ++d) { float c = cb[d]; dot += c * sF[d]; c2 += c * c; }
  sScore[tid] = c2 - 2.f * dot;        // ||f||^2 constant per token -> omitted
  sIdx[tid] = tid;
  __syncthreads();
  for (int s = 128; s > 0; s >>= 1) {  // argmin, first-index tie break
    if (tid < s) {
      float so = sScore[tid + s];
      int   io = sIdx[tid + s];
      if (so < sScore[tid] || (so == sScore[tid] && io < sIdx[tid])) {
        sScore[tid] = so; sIdx[tid] = io;
      }
    }
    __syncthreads();
  }
  const int best = sIdx[0];
  if (tid == 0) {
    idx_out[t] = best;
    out_indices[t] = (float)best;
    atomicAdd(&hist[best], 1);
  }
  __syncthreads();
  float d2 = 0.f;
  if (tid < EDIM) {
    float diff = codebook[(size_t)best * EDIM + tid] - sF[tid];
    d2 = diff * diff;
  }
  sScore[tid] = d2;
  __syncthreads();
  for (int s = 128; s > 0; s >>= 1) {
    if (tid < s) sScore[tid] += sScore[tid + s];
    __syncthreads();
  }
  if (tid == 0) atomicAdd(commit_accum, sScore[0]);
}

__global__ void vqvae_init(int* hist, float* commit) {
  hist[threadIdx.x] = 0;
  if (threadIdx.x == 0) commit[0] = 0.f;
}

__global__ __launch_bounds__(256) void vqvae_finalize(
    const int* __restrict__ hist, const float* __restrict__ commit_accum,
    float* __restrict__ out)
{
  __shared__ float red[256];
  const int tid = threadIdx.x;
  float p = (float)hist[tid] * (1.0f / (float)NTOK);
  red[tid] = p * logf(p + 1e-10f);
  __syncthreads();
  for (int s = 128; s > 0; s >>= 1) {
    if (tid < s) red[tid] += red[tid + s];
    __syncthreads();
  }
  if (tid == 0) {
    out[OFF_PERP]   = expf(-red[0]);
    out[OFF_COMMIT] = commit_accum[0] * (1.0f / (float)(NTOK * EDIM));
  }
}

// ---------------------------------------------------------------------------
// Kernel 4: decoder GEMMs. q_st == codebook[idx] exactly (straight-through).
// 32 blocks x 64 tokens; same f32 WMMA tiling as the encoder.
// ---------------------------------------------------------------------------
__global__ __launch_bounds__(256) void vqvae_decode(
    const int* __restrict__ idx, const float* __restrict__ codebook,
    const float* __restrict__ dw1, const float* __restrict__ db1,
    const float* __restrict__ dw2, const float* __restrict__ db2,
    float* __restrict__ dec_out)
{
  __shared__ float sQ[64 * SQ_STRIDE];
  __shared__ float sH[64 * SH_STRIDE];
  const int tid  = threadIdx.x;
  const int wave = tid >> 5;
  const int lane = tid & 31;
  const int tok0 = blockIdx.x * 64;

  for (int i = 0; i < 16; ++i) {       // 64 tokens x 64 dims
    int e = tid + (i << 8);
    int r = e >> 6, d = e & 63;
    sQ[r * SQ_STRIDE + d] = codebook[(size_t)idx[tok0 + r] * EDIM + d];
  }
  __syncthreads();

  const int n    = lane & 15;
  const int khi  = (lane >> 4) << 1;
  const int hi8  = (lane >> 4) << 3;
  const int col1 = (wave << 4) + n;

  v8f acc[4];
  #pragma unroll
  for (int t = 0; t < 4; ++t)
    acc[t] = (v8f){0.f,0.f,0.f,0.f,0.f,0.f,0.f,0.f};

  for (int kk = 0; kk < EDIM; kk += 4) {   // q[64x64] @ dw1[64x128]
    const int k0 = kk + khi;
    v2f bf;
    bf.x = dw1[k0 * E1 + col1];
    bf.y = dw1[(k0 + 1) * E1 + col1];
    #pragma unroll
    for (int t = 0; t < 4; ++t) {
      v2f af;
      af.x = sQ[(t * 16 + n) * SQ_STRIDE + k0];
      af.y = sQ[(t * 16 + n) * SQ_STRIDE + k0 + 1];
      acc[t] = wmma_k4(af, bf, acc[t]);
    }
  }
  const float bias1 = db1[col1];
  #pragma unroll
  for (int t = 0; t < 4; ++t)
    #pragma unroll
    for (int i = 0; i < 8; ++i)
      sH[(t * 16 + hi8 + i) * SH_STRIDE + col1] =
          gelu_exact(acc[t][i] + bias1);
  __syncthreads();

  // h[64x128] @ dw2[128x512]: 4x32 = 128 tiles, 16 per wave
  for (int tt = 0; tt < 16; ++tt) {
    const int id = wave + (tt << 3);
    const int mt = id >> 5;
    const int ct = id & 31;
    const int col2 = (ct << 4) + n;
    v8f a2 = (v8f){0.f,0.f,0.f,0.f,0.f,0.f,0.f,0.f};
    for (int kk = 0; kk < E1; kk += 4) {
      const int k0 = kk + khi;
      v2f bf;
      bf.x = dw2[k0 * DIN + col2];
      bf.y = dw2[(k0 + 1) * DIN + col2];
      v2f af;
      af.x = sH[(mt * 16 + n) * SH_STRIDE + k0];
      af.y = sH[(mt * 16 + n) * SH_STRIDE + k0 + 1];
      a2 = wmma_k4(af, bf, a2);
    }
    const float bias2 = db2[col2];
    #pragma unroll
    for (int i = 0; i < 8; ++i) {
      int row = mt * 16 + hi8 + i;
      dec_out[(size_t)(tok0 + row) * DIN + col2] = a2[i] + bias2;
    }
  }
}

// ---------------------------------------------------------------------------
// Kernel 5: bilinear 2x2 -> 14x14 (half-pixel, edge-clamped).  Block per batch,
// source slice in LDS, non-temporal streaming stores of the 205 MB output.
// ---------------------------------------------------------------------------
__global__ __launch_bounds__(256) void vqvae_upsample(
    const float* __restrict__ dec, float* __restrict__ out)
{
  __shared__ float s[4 * DIN];
  const int b = blockIdx.x, tid = threadIdx.x;
  for (int i = 0; i < 8; ++i) {
    int e = tid + (i << 8);
    s[e] = dec[(size_t)b * 4 * DIN + e];
  }
  __syncthreads();
  for (int i = 0; i < 392; ++i) {      // 14*14*512 / 256
    int e = tid + i * 256;
    int c = e & 511;
    int p = e >> 9;                    // 0..195
    int h = p / 14, w = p % 14;
    float th = fminf(fmaxf((h + 0.5f) * (1.0f / 7.0f) - 0.5f, 0.f), 1.f);
    float tw = fminf(fmaxf((w + 0.5f) * (1.0f / 7.0f) - 0.5f, 0.f), 1.f);
    float top = s[0 * DIN + c] * (1.f - tw) + s[1 * DIN + c] * tw;
    float bot = s[2 * DIN + c] * (1.f - tw) + s[3 * DIN + c] * tw;
    float v = top * (1.f - th) + bot * th;
    __builtin_nontemporal_store(v, out + (size_t)b * (H_N * W_N * DIN) + e);
  }
}

// ---------------------------------------------------------------------------
extern "C" void kernel_launch(void* const* d_in, const int* in_sizes, int n_in,
                              void* d_out, int out_size, void* d_ws, size_t ws_size,
                              hipStream_t stream) {
  (void)in_sizes; (void)n_in; (void)out_size; (void)ws_size;
  const float* t0  = (const float*)d_in[0];
  const float* t1  = (const float*)d_in[1];
  const float* ew1 = (const float*)d_in[2];
  const float* eb1 = (const float*)d_in[3];
  const float* ew2 = (const float*)d_in[4];
  const float* eb2 = (const float*)d_in[5];
  const float* dw1 = (const float*)d_in[6];
  const float* db1 = (const float*)d_in[7];
  const float* dw2 = (const float*)d_in[8];
  const float* db2 = (const float*)d_in[9];
  const float* cb  = (const float*)d_in[10];
  float* out = (float*)d_out;

  char* ws = (char*)d_ws;
  float* enc    = (float*)ws;                                  // 2048*64 f32
  float* dec    = (float*)(ws + 524288);                       // 2048*512 f32
  int*   idx    = (int*)  (ws + 524288 + 4194304);             // 2048 i32
  int*   hist   = (int*)  (ws + 524288 + 4194304 + 8192);      // 256 i32
  float* commit = (float*)(ws + 524288 + 4194304 + 8192 + 1024);

  vqvae_init<<<1, 256, 0, stream>>>(hist, commit);
  vqvae_encode_pool<<<NTOK, 256, 0, stream>>>(t0, t1, ew1, eb1, ew2, eb2, enc);
  vqvae_vq<<<NTOK, 256, 0, stream>>>(enc, cb, idx, commit, hist, out + OFF_IDX);
  vqvae_finalize<<<1, 256, 0, stream>>>(hist, commit, out);
  vqvae_decode<<<NTOK / 64, 256, 0, stream>>>(idx, cb, dw1, db1, dw2, db2, dec);
  vqvae_upsample<<<B_N, 256, 0, stream>>>(dec, out);
}